// ODEForecastModelGraph_34815004901553
// MI455X (gfx1250) — compile-verified
//
#include <hip/hip_runtime.h>
#include <hip/hip_bf16.h>
#include <math.h>

// Problem geometry (matches reference)
#define CDIM   256          // channel dim
#define TOUT   12           // output time steps
#define NDIM   2048
#define BDIM   8
#define NROWS_TOTAL (BDIM * NDIM)   // 16384 independent ODE rows

// Tiling
#define ROWS   64           // rows per workgroup
#define NWG    (NROWS_TOTAL / ROWS) // 256 workgroups
#define LDA    260          // LDS row stride (260 % 64 == 4 -> conflict-free A loads)
#define NSUB   4            // RK4 substeps per output interval
#define LN_EPS 1e-5f

typedef __attribute__((ext_vector_type(2))) float v2f;
typedef __attribute__((ext_vector_type(8))) float v8f;

// ---------------------------------------------------------------------------
// Wave-level GEMM via V_WMMA_F32_16X16X4_F32.
// Each of the 8 waves owns 2 UNIQUE column tiles (c0 = wave*32) and sweeps all
// 4 row strips -> B (global) loads are never duplicated across waves; the
// redundant operand (A, 8x reuse across waves) is served from LDS instead.
//
// fp32 WMMA operand layout (ISA 7.12.2):
//   A 16x4 : lanes 0-15 hold K=k0,k0+1 (vgpr0/1), lanes 16-31 hold K=k0+2,k0+3
//   B 4x16 : vgpr j, lanes 0-15 -> K=k0+j ; lanes 16-31 -> K=k0+2+j
//   C/D    : vgpr j, lanes 0-15 -> M=j    ; lanes 16-31 -> M=j+8
// ---------------------------------------------------------------------------
__device__ __forceinline__ void gemm_acc(const float* __restrict__ A,
                                         const float* __restrict__ Bg,
                                         const float* __restrict__ bias,
                                         int c0, int half, int ln16,
                                         v8f acc[4][2])
{
#pragma unroll
    for (int t = 0; t < 2; ++t) {
        float bv = bias[c0 + 16 * t + ln16];
#pragma unroll
        for (int rt = 0; rt < 4; ++rt)
            acc[rt][t] = (v8f){bv, bv, bv, bv, bv, bv, bv, bv};
    }
    const float* abase = A + ln16 * LDA;
#pragma unroll 2
    for (int k0 = 0; k0 < CDIM; k0 += 4) {
        v2f a[4];
#pragma unroll
        for (int rt = 0; rt < 4; ++rt) {
            const float2 af =
                *(const float2*)(abase + (16 * rt) * LDA + k0 + 2 * half);
            a[rt].x = af.x; a[rt].y = af.y;
        }
        const float* bbase = Bg + (size_t)(k0 + 2 * half) * CDIM + c0 + ln16;
#pragma unroll
        for (int t = 0; t < 2; ++t) {
            v2f b;
            b.x = bbase[16 * t];
            b.y = bbase[CDIM + 16 * t];
#pragma unroll
            for (int rt = 0; rt < 4; ++rt) {
                acc[rt][t] = __builtin_amdgcn_wmma_f32_16x16x4_f32(
                    false, a[rt], false, b, (short)0, acc[rt][t], false, false);
            }
        }
    }
}

__device__ __forceinline__ void store_tiles_lds(float* __restrict__ D,
                                                const v8f acc[4][2],
                                                int c0, int half, int ln16)
{
#pragma unroll
    for (int rt = 0; rt < 4; ++rt) {
#pragma unroll
        for (int t = 0; t < 2; ++t) {
#pragma unroll
            for (int j = 0; j < 8; ++j) {
                D[(16 * rt + 8 * half + j) * LDA + c0 + 16 * t + ln16] =
                    acc[rt][t][j];
            }
        }
    }
}

// Decode GEMM: out[:, ti] = Y @ dec_w + dec_b, written straight to HBM.
__device__ __forceinline__ void decode_frame(const float* __restrict__ Y,
                                             const float* __restrict__ dec_w,
                                             const float* __restrict__ dec_b,
                                             float* __restrict__ out,
                                             int grow0, int ti,
                                             int c0, int half, int ln16)
{
    v8f a8[4][2];
    gemm_acc(Y, dec_w, dec_b, c0, half, ln16, a8);
#pragma unroll
    for (int rt = 0; rt < 4; ++rt) {
#pragma unroll
        for (int t = 0; t < 2; ++t) {
#pragma unroll
            for (int j = 0; j < 8; ++j) {
                int rr = 16 * rt + 8 * half + j;
                int g  = grow0 + rr;
                int b  = g >> 11;          // g / 2048
                int n  = g & 2047;         // g % 2048
                size_t off = ((size_t)((b * TOUT + ti) * NDIM + n) << 8)
                             + (size_t)(c0 + 16 * t + ln16);
                out[off] = a8[rt][t][j];
            }
        }
    }
}

// Per-stage epilogue: KRAW holds cur@W+b. Compute h = silu, rowwise LN,
// k = cur + LN(h); ACC += wk * k; CUR = Y + a_next * k (or final y update).
// 4 threads per row, 64-element chunks, LDS partial reduction for mean/var.
__device__ __forceinline__ void stage_epilogue(float* __restrict__ Y,
                                               float* __restrict__ CUR,
                                               float* __restrict__ ACC,
                                               float* __restrict__ KR,
                                               float* __restrict__ RS,
                                               float* __restrict__ RQ,
                                               const float* __restrict__ ln_g,
                                               const float* __restrict__ ln_b,
                                               int tid, float wk, float a_next,
                                               int last, float dt)
{
    const int r = tid >> 2;        // row 0..63
    const int q = tid & 3;         // quarter 0..3
    float* kp = KR + r * LDA + q * 64;

    float s1 = 0.0f, s2 = 0.0f;
#pragma unroll 8
    for (int c = 0; c < 64; ++c) {
        float v = kp[c];
        float s = v / (1.0f + __expf(-v));   // silu
        kp[c] = s;
        s1 += s;
        s2 += s * s;
    }
    RS[tid] = s1;
    RQ[tid] = s2;
    __syncthreads();

    const int rb = r << 2;
    float sum = RS[rb] + RS[rb + 1] + RS[rb + 2] + RS[rb + 3];
    float sq  = RQ[rb] + RQ[rb + 1] + RQ[rb + 2] + RQ[rb + 3];
    float mu   = sum * (1.0f / (float)CDIM);
    float var  = sq * (1.0f / (float)CDIM) - mu * mu;
    float rstd = rsqrtf(var + LN_EPS);

    float* cp = CUR + r * LDA + q * 64;
    float* yp = Y   + r * LDA + q * 64;
    float* ap = ACC + r * LDA + q * 64;
    const float* gp = ln_g + q * 64;
    const float* bp = ln_b + q * 64;

    if (!last) {
#pragma unroll 4
        for (int c = 0; c < 64; ++c) {
            float h   = kp[c];
            float lnv = (h - mu) * rstd * gp[c] + bp[c];
            float kv  = cp[c] + lnv;          // k = f(cur)
            ap[c] += wk * kv;
            cp[c] = yp[c] + a_next * kv;      // next stage input
        }
    } else {
#pragma unroll 4
        for (int c = 0; c < 64; ++c) {
            float h   = kp[c];
            float lnv = (h - mu) * rstd * gp[c] + bp[c];
            float kv  = cp[c] + lnv;          // k4
            float yn  = yp[c] + dt * (1.0f / 6.0f) * (ap[c] + kv);
            yp[c] = yn;
            cp[c] = yn;                        // cur = y for next step
            ap[c] = 0.0f;                      // reset accumulator
        }
    }
    __syncthreads();
}

// ---------------------------------------------------------------------------
// Persistent kernel: encode -> 11 intervals x NSUB RK4 steps -> decode frames.
// Each workgroup owns 64 independent ODE rows entirely in LDS.
// ---------------------------------------------------------------------------
extern "C" __global__ __launch_bounds__(256)
void ode_forecast_kernel(const float* __restrict__ x,
                         const float* __restrict__ enc_w,
                         const float* __restrict__ enc_b,
                         const float* __restrict__ f_w,
                         const float* __restrict__ f_b,
                         const float* __restrict__ ln_g,
                         const float* __restrict__ ln_b,
                         const float* __restrict__ dec_w,
                         const float* __restrict__ dec_b,
                         float* __restrict__ out)
{
    extern __shared__ float smem[];
    float* Y   = smem;                       // 64 x LDA
    float* CUR = Y   + ROWS * LDA;
    float* ACC = CUR + ROWS * LDA;
    float* KR  = ACC + ROWS * LDA;
    float* RS  = KR  + ROWS * LDA;           // 256 partial sums
    float* RQ  = RS + 256;                   // 256 partial sumsq

    const int tid   = threadIdx.x;
    const int lane  = tid & 31;
    const int w     = tid >> 5;
    const int half  = lane >> 4;
    const int ln16  = lane & 15;
    const int c0    = w * 32;                // wave's unique column pair
    const int grow0 = blockIdx.x * ROWS;

    // Warm L2 with the stage / decode weights (global_prefetch_b8).
    {
        const char* p1 = (const char*)f_w;
        const char* p2 = (const char*)dec_w;
        for (int off = tid * 128; off < CDIM * CDIM * 4; off += 256 * 128) {
            __builtin_prefetch(p1 + off, 0, 1);
            __builtin_prefetch(p2 + off, 0, 1);
        }
    }

    // Load x[:, -1] rows for this workgroup into CUR; zero ACC.
    for (int e = tid; e < ROWS * CDIM; e += 256) {
        int r = e >> 8, c = e & 255;
        int g = grow0 + r;
        int b = g >> 11, n = g & 2047;
        size_t xoff = ((size_t)((b * TOUT + (TOUT - 1)) * NDIM + n) << 8) + c;
        CUR[r * LDA + c] = x[xoff];
        ACC[r * LDA + c] = 0.0f;
    }
    __syncthreads();

    // Encoder: Y = CUR @ enc_w + enc_b
    {
        v8f a8[4][2];
        gemm_acc(CUR, enc_w, enc_b, c0, half, ln16, a8);
        store_tiles_lds(Y, a8, c0, half, ln16);
    }
    __syncthreads();

    // cur = y0
    for (int e = tid; e < ROWS * CDIM; e += 256) {
        int r = e >> 8, c = e & 255;
        CUR[r * LDA + c] = Y[r * LDA + c];
    }
    __syncthreads();

    // Frame 0 (t = 0)
    decode_frame(Y, dec_w, dec_b, out, grow0, 0, c0, half, ln16);

    const float DT = (1.0f / (float)(TOUT - 1)) / (float)NSUB;

    for (int itv = 0; itv < TOUT - 1; ++itv) {
        for (int s = 0; s < NSUB; ++s) {
            // ---- stage 1: k1 = f(cur) ----
            { v8f a8[4][2]; gemm_acc(CUR, f_w, f_b, c0, half, ln16, a8);
              store_tiles_lds(KR, a8, c0, half, ln16); }
            __syncthreads();
            stage_epilogue(Y, CUR, ACC, KR, RS, RQ, ln_g, ln_b, tid,
                           1.0f, 0.5f * DT, 0, DT);
            // ---- stage 2: k2 = f(y + dt/2 k1) ----
            { v8f a8[4][2]; gemm_acc(CUR, f_w, f_b, c0, half, ln16, a8);
              store_tiles_lds(KR, a8, c0, half, ln16); }
            __syncthreads();
            stage_epilogue(Y, CUR, ACC, KR, RS, RQ, ln_g, ln_b, tid,
                           2.0f, 0.5f * DT, 0, DT);
            // ---- stage 3: k3 = f(y + dt/2 k2) ----
            { v8f a8[4][2]; gemm_acc(CUR, f_w, f_b, c0, half, ln16, a8);
              store_tiles_lds(KR, a8, c0, half, ln16); }
            __syncthreads();
            stage_epilogue(Y, CUR, ACC, KR, RS, RQ, ln_g, ln_b, tid,
                           2.0f, DT, 0, DT);
            // ---- stage 4: k4 = f(y + dt k3); y += dt/6 (k1+2k2+2k3+k4) ----
            { v8f a8[4][2]; gemm_acc(CUR, f_w, f_b, c0, half, ln16, a8);
              store_tiles_lds(KR, a8, c0, half, ln16); }
            __syncthreads();
            stage_epilogue(Y, CUR, ACC, KR, RS, RQ, ln_g, ln_b, tid,
                           1.0f, 0.0f, 1, DT);
        }
        // Emit frame itv+1
        decode_frame(Y, dec_w, dec_b, out, grow0, itv + 1, c0, half, ln16);
    }
}

extern "C" void kernel_launch(void* const* d_in, const int* in_sizes, int n_in,
                              void* d_out, int out_size, void* d_ws, size_t ws_size,
                              hipStream_t stream) {
    const float* x     = (const float*)d_in[0];
    const float* enc_w = (const float*)d_in[1];
    const float* enc_b = (const float*)d_in[2];
    const float* f_w   = (const float*)d_in[3];
    const float* f_b   = (const float*)d_in[4];
    const float* ln_g  = (const float*)d_in[5];
    const float* ln_b  = (const float*)d_in[6];
    const float* dec_w = (const float*)d_in[7];
    const float* dec_b = (const float*)d_in[8];
    float* out = (float*)d_out;

    const size_t shmem = (size_t)(4 * ROWS * LDA + 512) * sizeof(float); // ~268 KB of 320 KB
    hipLaunchKernelGGL(ode_forecast_kernel, dim3(NWG), dim3(256), shmem, stream,
                       x, enc_w, enc_b, f_w, f_b, ln_g, ln_b, dec_w, dec_b, out);
}